// DFA_72670846648769
// MI455X (gfx1250) — compile-verified
//
#include <hip/hip_runtime.h>
#include <hip/hip_bf16.h>
#include <math.h>

typedef __attribute__((ext_vector_type(2))) float v2f;
typedef __attribute__((ext_vector_type(8))) float v8f;
typedef int v4i_ __attribute__((vector_size(16)));

#define NSTATES 1024
#define NBLOCKS 64
#define ROWS_PER_BLOCK 16
#define TPB 256
#define NWAVES 8
#define PAD 1028   // 1028 % 64 == 4 -> conflict-free A-fragment reads in LDS

#if __has_builtin(__builtin_amdgcn_global_load_async_to_lds_b128)
#define HAVE_ASYNC_LDS 1
#else
#define HAVE_ASYNC_LDS 0
#endif

// builtin wants typed v4i pointers: global (as1) source, LDS (as3) dest
#define GPTR(p) ((__attribute__((address_space(1))) v4i_*)(p))
#define LPTR(p) ((__attribute__((address_space(3))) v4i_*)(p))

// ---------------------------------------------------------------------------
// init: zero state buffers, set q0 = e_0 (p = e_0, S = 1)
// ---------------------------------------------------------------------------
__global__ void dfa_init(float* pbuf, float* spart, unsigned* sync) {
    int t = threadIdx.x;
    for (int j = t; j < 2 * NSTATES; j += TPB) pbuf[j] = 0.0f;
    for (int j = t; j < 2 * NBLOCKS; j += TPB) spart[j] = 0.0f;
    if (t == 0) {
        pbuf[0]  = 1.0f;   // p_prev = e_0
        spart[0] = 1.0f;   // S_prev = 1
        sync[0] = 0u;      // barrier arrival count
        sync[1] = 0u;      // barrier generation flag
    }
}

// ---------------------------------------------------------------------------
// stats: per (symbol, column) softmax statistics folded into one coefficient
//   cw[s][j] = exp(-5*mx) * se^-5 / cs   so that  X5[i][j] = exp(5*d[i][j])*cw[j]
// ---------------------------------------------------------------------------
__global__ void dfa_stats(const float* __restrict__ delta,
                          float* __restrict__ cw, int ncols) {
    int g = blockIdx.x * blockDim.x + threadIdx.x;
    if (g >= ncols) return;
    int s = g >> 10;
    int j = g & (NSTATES - 1);
    const float* col = delta + ((size_t)s << 20) + j;

    float mx = -INFINITY, se = 0.0f;
    for (int i = 0; i < NSTATES; ++i) {
        float v = col[(size_t)i << 10];
        if (v > mx) { se *= __expf(mx - v); mx = v; }
        se += __expf(v - mx);
    }
    float invse = 1.0f / se;
    float cs = 0.0f;   // faithful colsum of rounded softmax (~1.0)
    for (int i = 0; i < NSTATES; ++i)
        cs += __expf(col[(size_t)i << 10] - mx) * invse;

    float i2 = invse * invse;
    float i5 = i2 * i2 * invse;
    cw[g] = __expf(-5.0f * mx) * i5 / cs;
}

// ---------------------------------------------------------------------------
// grid-wide barrier (monotonic generation counter)
// ---------------------------------------------------------------------------
__device__ __forceinline__ void grid_barrier(unsigned* cnt, unsigned* flag,
                                             unsigned gen) {
    __threadfence();
    __syncthreads();
    if (threadIdx.x == 0) {
        unsigned prev = __hip_atomic_fetch_add(cnt, 1u, __ATOMIC_ACQ_REL,
                                               __HIP_MEMORY_SCOPE_AGENT);
        if (prev == (gen + 1u) * NBLOCKS - 1u) {
            __hip_atomic_store(flag, gen + 1u, __ATOMIC_RELEASE,
                               __HIP_MEMORY_SCOPE_AGENT);
        } else {
            while (__hip_atomic_load(flag, __ATOMIC_ACQUIRE,
                                     __HIP_MEMORY_SCOPE_AGENT) < gen + 1u)
                __builtin_amdgcn_s_sleep(2);
        }
    }
    __syncthreads();
    __threadfence();
}

// ---------------------------------------------------------------------------
// issue one tile (16 rows x 1024 f32, padded rows) as async DMA global->LDS
// ---------------------------------------------------------------------------
__device__ __forceinline__ void issue_tile_async(float* dst, const float* src) {
#if HAVE_ASYNC_LDS
    const int tid = threadIdx.x;
#pragma unroll
    for (int i = 0; i < (ROWS_PER_BLOCK * 256) / TPB; ++i) {   // 16 x B128/thread
        int idx = tid + i * TPB;
        int r  = idx >> 8;
        int c  = (idx & 255) << 2;                              // float index
        __builtin_amdgcn_global_load_async_to_lds_b128(
            GPTR(src + (size_t)r * NSTATES + c),
            LPTR(dst + r * PAD + c), 0, 0);
    }
#else
    (void)dst; (void)src;
#endif
}

__device__ __forceinline__ void wait_async_lds() {
#if HAVE_ASYNC_LDS
#if __has_builtin(__builtin_amdgcn_s_wait_asynccnt)
    __builtin_amdgcn_s_wait_asynccnt(0);
#else
    asm volatile("s_wait_asynccnt 0" ::: "memory");
#endif
#endif
}

// ---------------------------------------------------------------------------
// main sequential recurrence: one grid barrier per step, async-prefetched
// double-buffered tiles, 4-way accumulator f32 WMMA matvec
// ---------------------------------------------------------------------------
__global__ void __launch_bounds__(TPB, 1)
dfa_main(const int* __restrict__ seq, const float* __restrict__ delta,
         float* pbuf, float* spart, unsigned* sync,
         const float* __restrict__ cw, int T) {
    __shared__ float tiles[2][ROWS_PER_BLOCK * PAD];  // raw delta rows (2 bufs)
    __shared__ float wls[NSTATES];                    // per-column weights
    __shared__ float pc[NWAVES][16];
    __shared__ float red[NWAVES];
    __shared__ float prow[ROWS_PER_BLOCK];
    __shared__ float sq_sh;

    const int tid  = threadIdx.x;
    const int lane = tid & 31;
    const int wave = tid >> 5;
    const int row0 = blockIdx.x * ROWS_PER_BLOCK;
    const int rloc = lane & 15;          // A-fragment row
    const int koff = (lane >> 4) << 1;   // A-fragment K offset (lanes 16-31)

    unsigned* cnt  = sync;
    unsigned* flag = sync + 1;

#if HAVE_ASYNC_LDS
    // prime: DMA step-0 tile into buffer 0
    if (T > 0)
        issue_tile_async(tiles[0],
                         delta + ((size_t)seq[0] << 20) + (size_t)row0 * NSTATES);
#endif

    for (int t = 0; t < T; ++t) {
        const int par = t & 1;
        const float* pprev = pbuf + par * NSTATES;
        float*       pnext = pbuf + (par ^ 1) * NSTATES;
        const float* sprev = spart + par * NBLOCKS;
        float*       snext = spart + (par ^ 1) * NBLOCKS;

        const int sym = seq[t];
        const float* cwS = cw + (sym << 10);

        // ---- phase A: S = sum(p_prev); q = p/S; sq = sum(q); w = q^5/sq*cw
        float S = 0.0f;
        for (int b = 0; b < NBLOCKS; ++b) S += sprev[b];
        const float invS = 1.0f / S;

        float part = 0.0f;
        for (int j = tid; j < NSTATES; j += TPB) {
            float q = pprev[j] * invS;
            wls[j] = q;
            part += q;
        }
        for (int off = 16; off > 0; off >>= 1)
            part += __shfl_down(part, off, 32);
        if (lane == 0) red[wave] = part;
        __syncthreads();
        if (tid == 0) {
            float sq = 0.0f;
            for (int w2 = 0; w2 < NWAVES; ++w2) sq += red[w2];
            sq_sh = sq;
        }
        __syncthreads();
        const float inv_sq = 1.0f / sq_sh;
        for (int j = tid; j < NSTATES; j += TPB) {
            float q  = wls[j];
            float q2 = q * q;
            wls[j] = q2 * q2 * q * inv_sq * cwS[j];   // q^5/sq * cw
        }

#if HAVE_ASYNC_LDS
        // current tile's DMA (issued last step) must be complete and visible
        wait_async_lds();
        __syncthreads();
        // kick off next step's tile DMA into the other buffer (fully hidden
        // behind this step's WMMA + reduction + grid barrier)
        if (t + 1 < T)
            issue_tile_async(tiles[par ^ 1],
                             delta + ((size_t)seq[t + 1] << 20) +
                             (size_t)row0 * NSTATES);
#else
        // fallback: synchronous staging of the current tile (raw values)
        {
            const float* src = delta + ((size_t)sym << 20) +
                               (size_t)row0 * NSTATES;
            const int c0 = tid << 2;
#pragma unroll 4
            for (int r = 0; r < ROWS_PER_BLOCK; ++r) {
                const float4 dv = *(const float4*)(src + (size_t)r * NSTATES + c0);
                float* dst = &tiles[par][r * PAD + c0];
                dst[0] = dv.x; dst[1] = dv.y; dst[2] = dv.z; dst[3] = dv.w;
            }
        }
        __syncthreads();
#endif

        // warm L2 for step t+2 (global_prefetch_b8)
        if (t + 2 < T) {
            const char* nb = (const char*)(delta + ((size_t)seq[t + 2] << 20) +
                                           (size_t)row0 * NSTATES);
            const int nlines = (ROWS_PER_BLOCK * NSTATES * 4) / 128;   // 512
            for (int l = tid; l < nlines; l += TPB)
                __builtin_prefetch(nb + (size_t)l * 128, 0, 3);
        }

        // ---- phase C: WMMA f32 16x16x4 matvec, exp applied at fragment load,
        //      4 independent accumulators for XDL pipeline depth
        const float* tile = tiles[par];
        v8f c0 = {0.f,0.f,0.f,0.f,0.f,0.f,0.f,0.f};
        v8f c1 = c0, c2 = c0, c3 = c0;
        const int kstart =
            __builtin_amdgcn_readfirstlane(wave) * (NSTATES / NWAVES);  // 128
#pragma unroll 4
        for (int k = kstart; k < kstart + NSTATES / NWAVES; k += 16) {
            v2f r0 = *(const v2f*)&tile[rloc * PAD + k + koff];
            v2f r1 = *(const v2f*)&tile[rloc * PAD + k + 4 + koff];
            v2f r2 = *(const v2f*)&tile[rloc * PAD + k + 8 + koff];
            v2f r3 = *(const v2f*)&tile[rloc * PAD + k + 12 + koff];
            v2f b0 = *(const v2f*)&wls[k + koff];
            v2f b1 = *(const v2f*)&wls[k + 4 + koff];
            v2f b2 = *(const v2f*)&wls[k + 8 + koff];
            v2f b3 = *(const v2f*)&wls[k + 12 + koff];
            v2f a0, a1, a2, a3;
            a0.x = __expf(5.0f * r0.x); a0.y = __expf(5.0f * r0.y);
            a1.x = __expf(5.0f * r1.x); a1.y = __expf(5.0f * r1.y);
            a2.x = __expf(5.0f * r2.x); a2.y = __expf(5.0f * r2.y);
            a3.x = __expf(5.0f * r3.x); a3.y = __expf(5.0f * r3.y);
            c0 = __builtin_amdgcn_wmma_f32_16x16x4_f32(
                     false, a0, false, b0, (short)0, c0, false, false);
            c1 = __builtin_amdgcn_wmma_f32_16x16x4_f32(
                     false, a1, false, b1, (short)0, c1, false, false);
            c2 = __builtin_amdgcn_wmma_f32_16x16x4_f32(
                     false, a2, false, b2, (short)0, c2, false, false);
            c3 = __builtin_amdgcn_wmma_f32_16x16x4_f32(
                     false, a3, false, b3, (short)0, c3, false, false);
        }
        v8f c = (c0 + c1) + (c2 + c3);

        // D col 0: rows 0-7 in lane 0 c[0..7], rows 8-15 in lane 16 c[0..7]
        if (lane == 0) {
#pragma unroll
            for (int v = 0; v < 8; ++v) pc[wave][v] = c[v];
        } else if (lane == 16) {
#pragma unroll
            for (int v = 0; v < 8; ++v) pc[wave][8 + v] = c[v];
        }
        __syncthreads();

        if (tid < ROWS_PER_BLOCK) {
            float s = 0.0f;
            for (int w2 = 0; w2 < NWAVES; ++w2) s += pc[w2][tid];
            pnext[row0 + tid] = s;
            prow[tid] = s;
        }
        __syncthreads();
        if (tid == 0) {
            float bs = 0.0f;
            for (int r = 0; r < ROWS_PER_BLOCK; ++r) bs += prow[r];
            snext[blockIdx.x] = bs;
        }

        grid_barrier(cnt, flag, (unsigned)t);
    }
}

// ---------------------------------------------------------------------------
// final: out = (p/sum(p)) . f
// ---------------------------------------------------------------------------
__global__ void dfa_final(const float* __restrict__ p,
                          const float* __restrict__ spart,
                          const float* __restrict__ f, float* out) {
    __shared__ float red[NWAVES];
    const int tid = threadIdx.x, lane = tid & 31, wave = tid >> 5;
    float S = 0.0f;
    for (int b = 0; b < NBLOCKS; ++b) S += spart[b];
    const float invS = 1.0f / S;
    float acc = 0.0f;
    for (int j = tid; j < NSTATES; j += TPB) acc += (p[j] * invS) * f[j];
    for (int off = 16; off > 0; off >>= 1) acc += __shfl_down(acc, off, 32);
    if (lane == 0) red[wave] = acc;
    __syncthreads();
    if (tid == 0) {
        float s = 0.0f;
        for (int w2 = 0; w2 < NWAVES; ++w2) s += red[w2];
        out[0] = s;
    }
}

// ---------------------------------------------------------------------------
extern "C" void kernel_launch(void* const* d_in, const int* in_sizes, int n_in,
                              void* d_out, int out_size, void* d_ws, size_t ws_size,
                              hipStream_t stream) {
    (void)n_in; (void)out_size; (void)ws_size;
    const int*   seq   = (const int*)d_in[0];
    const float* delta = (const float*)d_in[1];
    const float* f     = (const float*)d_in[2];
    const int T = in_sizes[0];
    const int A = in_sizes[1] / (NSTATES * NSTATES);   // 64 symbols

    char* ws = (char*)d_ws;
    float*    pbuf  = (float*)ws;                               // 2*1024 f32
    float*    spart = (float*)(ws + 2 * NSTATES * 4);           // 2*64 f32
    unsigned* sync  = (unsigned*)(ws + 2 * NSTATES * 4 + 2 * NBLOCKS * 4);
    float*    cw    = (float*)(ws + 9216);                      // A*1024 f32

    dfa_init<<<1, TPB, 0, stream>>>(pbuf, spart, sync);
    dfa_stats<<<(A * NSTATES + TPB - 1) / TPB, TPB, 0, stream>>>(delta, cw,
                                                                 A * NSTATES);
    dfa_main<<<NBLOCKS, TPB, 0, stream>>>(seq, delta, pbuf, spart, sync, cw, T);

    const int fpar = T & 1;
    dfa_final<<<1, TPB, 0, stream>>>(pbuf + fpar * NSTATES,
                                     spart + fpar * NBLOCKS, f, (float*)d_out);
}